// SparseMaskNetwork_8358006358542
// MI455X (gfx1250) — compile-verified
//
#include <hip/hip_runtime.h>
#include <hip/hip_bf16.h>

typedef float v2f __attribute__((ext_vector_type(2)));
typedef float v8f __attribute__((ext_vector_type(8)));
typedef int   v4i __attribute__((__vector_size__(16)));   // matches builtin param
typedef __attribute__((address_space(1))) v4i gv4i;       // global
typedef __attribute__((address_space(3))) v4i lv4i;       // LDS

#define TILE    256
#define FSTRIDE 104   // 96 channels padded; 2-voxel offset = 208 dwords = +16 banks
#define MAXM    64

#if defined(__has_builtin)
# if __has_builtin(__builtin_amdgcn_global_load_async_to_lds_b128)
#  define HAVE_ASYNC_B128 1
# else
#  define HAVE_ASYNC_B128 0
# endif
# if __has_builtin(__builtin_amdgcn_s_wait_asynccnt)
#  define HAVE_WAIT_ASYNC 1
# else
#  define HAVE_WAIT_ASYNC 0
# endif
#else
# define HAVE_ASYNC_B128 0
# define HAVE_WAIT_ASYNC 0
#endif

__global__ __launch_bounds__(256)
void zero_roi_kernel(float* __restrict__ out, int n) {
    int i = blockIdx.x * blockDim.x + threadIdx.x;
    if (i < n) out[i] = 0.0f;
}

__global__ __launch_bounds__(256)
void roi_wmma_kernel(const int* __restrict__ coords,
                     const int* __restrict__ batch_idx,
                     const float* __restrict__ features,
                     const float* __restrict__ conv_features,
                     const int* __restrict__ box_batch,
                     const int* __restrict__ box_min,
                     const int* __restrict__ box_max,
                     float* __restrict__ out,
                     int N, int M)
{
    extern __shared__ char smem[];
    float*    sFeat   = (float*)smem;                        // [TILE][FSTRIDE]
    unsigned* sMaskLo = (unsigned*)(sFeat + TILE * FSTRIDE); // [TILE]
    unsigned* sMaskHi = sMaskLo + TILE;                      // [TILE]
    unsigned* sGrpLo  = sMaskHi + TILE;                      // [TILE/4]
    unsigned* sGrpHi  = sGrpLo + (TILE / 4);                 // [TILE/4]

    const int tid  = (int)threadIdx.x;
    const int lane = tid & 31;
    const int wave = tid >> 5;

    // wave-constant tile assignment, forced into SGPRs so the group-skip
    // below is a pure scalar branch (s_and / s_cmp / s_cbranch).
    const int bTile = __builtin_amdgcn_readfirstlane(wave & 3);  // 16-box tile
    const int cHalf = __builtin_amdgcn_readfirstlane(wave >> 2); // 48-channel half
    const int cBase = cHalf * 48;
    const unsigned grpSel = 0xFFFFu << ((bTile & 1) * 16);

    const size_t skipOff = (size_t)M * 96;
    const size_t selOff  = skipOff + (size_t)N * 32;

    v8f acc0 = {}; v8f acc1 = {}; v8f acc2 = {};

    const int laneHalf = lane >> 4;
    const int laneLo   = lane & 15;

    const int numTiles = (N + TILE - 1) / TILE;
    for (int t = blockIdx.x; t < numTiles; t += gridDim.x) {
        __syncthreads();                        // previous phase B done with LDS
        if (tid < TILE / 4) { sGrpLo[tid] = 0u; sGrpHi[tid] = 0u; }
        __syncthreads();

        // ---------------- Phase A: stream voxels, masks, skip/selection --------
        const int v  = t * TILE + tid;
        const int vc = (v < N) ? v : (N - 1);   // clamped: uniform control flow
        float* dstF = sFeat + tid * FSTRIDE;

        // conv features -> LDS channels [0..63].
#if HAVE_ASYNC_B128
        {
            gv4i* gsrc = (gv4i*)(conv_features + (size_t)vc * 64);
            lv4i* ldst = (lv4i*)dstF;
            #pragma unroll
            for (int i = 0; i < 16; ++i)
                __builtin_amdgcn_global_load_async_to_lds_b128(
                    gsrc + i, ldst + i, 0, 0);
        }
#else
        {
            const float4* cf = (const float4*)(conv_features + (size_t)vc * 64);
            float4* dc = (float4*)dstF;
            #pragma unroll
            for (int i = 0; i < 16; ++i) dc[i] = cf[i];
        }
#endif

        // prefetch next tile's feature lines for this thread
        {
            const int vn = vc + (int)gridDim.x * TILE;
            if (vn < N) {
                __builtin_prefetch(conv_features + (size_t)vn * 64, 0, 0);
                __builtin_prefetch(features + (size_t)vn * 32, 0, 0);
            }
        }

        // box test: box descriptors read with uniform index in uniform control
        // flow -> scalarized to s_load; compares are VALU-vs-SGPR.
        const int b = batch_idx[vc];
        const int x = coords[(size_t)vc * 3 + 0];
        const int y = coords[(size_t)vc * 3 + 1];
        const int z = coords[(size_t)vc * 3 + 2];
        unsigned mLo = 0u, mHi = 0u;
        #pragma unroll 8
        for (int m = 0; m < MAXM; ++m) {
            const int bb  = box_batch[m];
            const int mnx = box_min[m * 3 + 0], mxx = box_max[m * 3 + 0];
            const int mny = box_min[m * 3 + 1], mxy = box_max[m * 3 + 1];
            const int mnz = box_min[m * 3 + 2], mxz = box_max[m * 3 + 2];
            const unsigned in =
                (unsigned)((bb == b) &
                           (x >= mnx) & (x < mxx) &
                           (y >= mny) & (y < mxy) &
                           (z >= mnz) & (z < mxz));
            if (m < 32) mLo |= in << m; else mHi |= in << (m - 32);
        }
        if (v >= N) { mLo = 0u; mHi = 0u; }     // duplicate voxel contributes 0
        const float sel = (mLo | mHi) ? 1.0f : 0.0f;

        // raw features -> LDS channels [64..95]; masked skip + selection writes
        {
            const float4* rf = (const float4*)(features + (size_t)vc * 32);
            float4* dr = (float4*)(dstF + 64);
            float4 q[8];
            #pragma unroll
            for (int i = 0; i < 8; ++i) { q[i] = rf[i]; dr[i] = q[i]; }
            if (v < N) {
                out[selOff + (size_t)v] = sel;
                float4* skp = (float4*)(out + skipOff + (size_t)v * 32);
                #pragma unroll
                for (int i = 0; i < 8; ++i) {
                    float4 s;
                    s.x = q[i].x * sel; s.y = q[i].y * sel;
                    s.z = q[i].z * sel; s.w = q[i].w * sel;
                    skp[i] = s;
                }
            }
        }

        sMaskLo[tid] = mLo;
        sMaskHi[tid] = mHi;
        if (mLo) atomicOr(&sGrpLo[tid >> 2], mLo);
        if (mHi) atomicOr(&sGrpHi[tid >> 2], mHi);

#if HAVE_ASYNC_B128
# if HAVE_WAIT_ASYNC
        __builtin_amdgcn_s_wait_asynccnt(0);
# else
        asm volatile("s_wait_asynccnt 0x0" ::: "memory");
# endif
#endif
        __syncthreads();

        // ---------------- Phase B: WMMA f32 16x16x4 masked accumulation --------
        for (int k = 0; k < TILE; k += 4) {
            const int g = k >> 2;
            // wave-uniform scalar skip; EXEC stays all-1s at every WMMA
            const unsigned go = (unsigned)__builtin_amdgcn_readfirstlane(
                (int)((bTile < 2) ? sGrpLo[g] : sGrpHi[g]));
            if ((go & grpSel) == 0u) continue;

            const int vox0 = k + laneHalf * 2;
            const int vox1 = vox0 + 1;
            const unsigned w0 = (bTile < 2) ? sMaskLo[vox0] : sMaskHi[vox0];
            const unsigned w1 = (bTile < 2) ? sMaskLo[vox1] : sMaskHi[vox1];
            const int shift = (bTile & 1) * 16 + laneLo;

            v2f a;
            a.x = ((w0 >> shift) & 1u) ? 1.0f : 0.0f;
            a.y = ((w1 >> shift) & 1u) ? 1.0f : 0.0f;

            const float* f0 = sFeat + vox0 * FSTRIDE + cBase + laneLo;
            const float* f1 = sFeat + vox1 * FSTRIDE + cBase + laneLo;
            v2f b0; b0.x = f0[0];  b0.y = f1[0];
            v2f b1; b1.x = f0[16]; b1.y = f1[16];
            v2f b2; b2.x = f0[32]; b2.y = f1[32];

            acc0 = __builtin_amdgcn_wmma_f32_16x16x4_f32(false, a, false, b0,
                                                         (short)0, acc0, false, false);
            acc1 = __builtin_amdgcn_wmma_f32_16x16x4_f32(false, a, false, b1,
                                                         (short)0, acc1, false, false);
            acc2 = __builtin_amdgcn_wmma_f32_16x16x4_f32(false, a, false, b2,
                                                         (short)0, acc2, false, false);
        }
    }

    // ---------------- Flush accumulators: one atomic contribution per block ----
    const int boxBase = bTile * 16 + laneHalf * 8;
    #pragma unroll
    for (int r = 0; r < 8; ++r) {
        const int box = boxBase + r;
        if (box < M) {
            atomicAdd(&out[(size_t)box * 96 + cBase +  0 + laneLo], acc0[r]);
            atomicAdd(&out[(size_t)box * 96 + cBase + 16 + laneLo], acc1[r]);
            atomicAdd(&out[(size_t)box * 96 + cBase + 32 + laneLo], acc2[r]);
        }
    }
}

extern "C" void kernel_launch(void* const* d_in, const int* in_sizes, int n_in,
                              void* d_out, int out_size, void* d_ws, size_t ws_size,
                              hipStream_t stream) {
    const int*   coords        = (const int*)d_in[0];
    const int*   batch_idx     = (const int*)d_in[1];
    const float* features      = (const float*)d_in[2];
    const float* conv_features = (const float*)d_in[3];
    const int*   box_batch     = (const int*)d_in[4];
    const int*   box_min       = (const int*)d_in[5];
    const int*   box_max       = (const int*)d_in[6];
    float*       out           = (float*)d_out;

    const int N = in_sizes[1];   // batch_idx element count
    const int M = in_sizes[4];   // box count (64)

    const int roiElems = M * 96;
    zero_roi_kernel<<<(roiElems + 255) / 256, 256, 0, stream>>>(out, roiElems);

    const int numTiles = (N + TILE - 1) / TILE;
    int blocks = numTiles < 512 ? numTiles : 512;

    const size_t shmem =
        (size_t)TILE * FSTRIDE * sizeof(float) +     // staged combined features
        (size_t)TILE * 2 * sizeof(unsigned) +        // per-voxel 64-bit masks
        (size_t)(TILE / 4) * 2 * sizeof(unsigned);   // per-4-voxel group OR

    roi_wmma_kernel<<<blocks, 256, shmem, stream>>>(
        coords, batch_idx, features, conv_features,
        box_batch, box_min, box_max, out, N, M);
}